// GroupedConvFuseSide4_18992345383204
// MI455X (gfx1250) — compile-verified
//
#include <hip/hip_runtime.h>

// Streaming fused grouped-1x1-conv:
//   out[b,k,h,w] = w[k,0]*side5[b,k] + w[k,1]*side4[b,k]
//                + w[k,2]*side1[b,0] + w[k,3]*side2[b,0] + w[k,4]*side3[b,0] + bias[k]
//
// Roofline: ~480 MiB HBM traffic, ~0.4 GFLOP -> 0.9 flop/byte; deeply
// memory-bound, so this is a pure b128 stream kernel. Each thread owns one
// float4 pixel-group for ALL 19 classes, so side1/2/3 are fetched from HBM
// exactly once and reused from registers. side4/side5/out use non-temporal
// hints (each stream is 152 MiB; NT keeps them from thrashing the 192 MB L2
// against each other). Weights/bias are uniform -> scalar cache.
//
// K=19: manually paired k+=2 loop (9 iterations, straight-line body, no
// remainder conditional) + one peeled tail. #pragma unroll 1 pins the loop
// rolled — full unroll (round 3) made the compiler hoist the whole 114-float
// weight table into SGPRs and spill via v_writelane (526 VALU instructions).

typedef float v4f __attribute__((ext_vector_type(4)));

__device__ __forceinline__ v4f fused_fma(const v4f a1, const v4f a2, const v4f a3,
                                         const v4f x4, const v4f x5,
                                         float w0, float w1, float w2,
                                         float w3, float w4, float bk)
{
    v4f r;
#pragma unroll
    for (int i = 0; i < 4; ++i) {
        float acc = __builtin_fmaf(a1[i], w2, bk);
        acc = __builtin_fmaf(a2[i], w3, acc);
        acc = __builtin_fmaf(a3[i], w4, acc);
        acc = __builtin_fmaf(x4[i], w1, acc);
        acc = __builtin_fmaf(x5[i], w0, acc);
        r[i] = acc;
    }
    return r;
}

__global__ __launch_bounds__(256) void grouped_conv_stream_kernel(
    const float* __restrict__ s1, const float* __restrict__ s2,
    const float* __restrict__ s3, const float* __restrict__ s4,
    const float* __restrict__ s5, const float* __restrict__ w,
    const float* __restrict__ bias, float* __restrict__ out)
{
    constexpr int K  = 19;
    constexpr unsigned HW = 512u * 512u;   // per-channel plane (elements)
    constexpr unsigned P4 = HW / 4u;       // float4 groups per plane
    constexpr unsigned NPIX4 = 8u * P4;    // total float4 groups over (b,h,w)

    const unsigned gid = blockIdx.x * blockDim.x + threadIdx.x;
    if (gid >= NPIX4) return;

    const unsigned b  = gid >> 16;          // gid / P4   (P4 == 65536)
    const unsigned p4 = gid & (P4 - 1u);    // gid % P4
    const unsigned sOff = b * HW + p4 * 4u;

    // Singles: fetched once, reused for all 19 classes from registers.
    const v4f a1 = __builtin_nontemporal_load((const v4f*)(s1 + sOff));
    const v4f a2 = __builtin_nontemporal_load((const v4f*)(s2 + sOff));
    const v4f a3 = __builtin_nontemporal_load((const v4f*)(s3 + sOff));

    // 32-bit offset: max element offset = 8*19*262144 < 2^31, so a single
    // 32-bit add per iteration suffices (avoids 64-bit addr-math chains).
    unsigned off = b * (K * HW) + p4 * 4u;

    // Main loop: 9 rolled, straight-line double-iterations (k = 0..17).
#pragma unroll 1
    for (int k = 0; k < K - 1; k += 2, off += 2u * HW) {
        const unsigned offA = off;
        const unsigned offB = off + HW;

        const v4f x5a = __builtin_nontemporal_load((const v4f*)(s5 + offA));
        const v4f x4a = __builtin_nontemporal_load((const v4f*)(s4 + offA));
        const v4f x5b = __builtin_nontemporal_load((const v4f*)(s5 + offB));
        const v4f x4b = __builtin_nontemporal_load((const v4f*)(s4 + offB));

        const int ka = k, kb = k + 1;
        const v4f ra = fused_fma(a1, a2, a3, x4a, x5a,
                                 w[ka * 5 + 0], w[ka * 5 + 1], w[ka * 5 + 2],
                                 w[ka * 5 + 3], w[ka * 5 + 4], bias[ka]);
        const v4f rb = fused_fma(a1, a2, a3, x4b, x5b,
                                 w[kb * 5 + 0], w[kb * 5 + 1], w[kb * 5 + 2],
                                 w[kb * 5 + 3], w[kb * 5 + 4], bias[kb]);

        __builtin_nontemporal_store(ra, (v4f*)(out + offA));
        __builtin_nontemporal_store(rb, (v4f*)(out + offB));
    }

    // Peeled tail: k = 18 (compile-time constant offsets into w/bias).
    {
        constexpr int k = K - 1;
        const v4f x5 = __builtin_nontemporal_load((const v4f*)(s5 + off));
        const v4f x4 = __builtin_nontemporal_load((const v4f*)(s4 + off));
        const v4f r  = fused_fma(a1, a2, a3, x4, x5,
                                 w[k * 5 + 0], w[k * 5 + 1], w[k * 5 + 2],
                                 w[k * 5 + 3], w[k * 5 + 4], bias[k]);
        __builtin_nontemporal_store(r, (v4f*)(out + off));
    }
}

extern "C" void kernel_launch(void* const* d_in, const int* in_sizes, int n_in,
                              void* d_out, int out_size, void* d_ws, size_t ws_size,
                              hipStream_t stream) {
    (void)in_sizes; (void)n_in; (void)d_ws; (void)ws_size; (void)out_size;

    const float* s1   = (const float*)d_in[0];
    const float* s2   = (const float*)d_in[1];
    const float* s3   = (const float*)d_in[2];
    const float* s4   = (const float*)d_in[3];
    const float* s5   = (const float*)d_in[4];
    const float* w    = (const float*)d_in[5];
    const float* bias = (const float*)d_in[6];
    float* out        = (float*)d_out;

    constexpr int NPIX4 = 8 * (512 * 512) / 4;   // 524288 float4 groups
    constexpr int BLOCK = 256;                   // 8 wave32s per workgroup
    const int grid = (NPIX4 + BLOCK - 1) / BLOCK;

    grouped_conv_stream_kernel<<<grid, BLOCK, 0, stream>>>(
        s1, s2, s3, s4, s5, w, bias, out);
}